// Gate_69887707840707
// MI455X (gfx1250) — compile-verified
//
#include <hip/hip_runtime.h>

// f32 WMMA operand vectors (per lane, wave32)
typedef __attribute__((ext_vector_type(2))) float v2f;  // A or B operand: 64 f32 / 32 lanes
typedef __attribute__((ext_vector_type(8))) float v8f;  // C/D: 256 f32 / 32 lanes

#define C6_4 1296
#define C6_5 7776
#define C6_6 46656
#define C6_7 279936
#define C6_8 1679616
#define C6_9 10077696
#define NUM_TILES 104976   // (6^8 columns) / 16 cols-per-tile = 1296 * 81

// Out[AB, n] = sum_ab Bmat[AB, ab] * X[ab, n]
//   row offset(r) = (r/6)*6^7 + (r%6)*6^4  (gate axes 2 and 5 of the 6^10 tensor)
//   col n = rest_outer(i0,i1,i3,i4 strides 6^9,6^8,6^6,6^5) + contiguous rest_inner (1296)
__global__ void __launch_bounds__(128, 1)
gate_wmma_kernel(const float* __restrict__ X,
                 const float* __restrict__ Bm,
                 float* __restrict__ Out) {
    // Gate matrix in LDS, padded to 48 rows (rows 36..47 = 0) so the M-padding
    // needs no branches and A operands are plain ds_load_b64 fetches.
    __shared__ float lgate[48 * 36];   // 6912 B
    for (int i = threadIdx.x; i < 48 * 36; i += 128)
        lgate[i] = (i < 36 * 36) ? Bm[i] : 0.0f;
    __syncthreads();

    const int lane   = threadIdx.x & 31;
    const int lane_n = lane & 15;   // column within 16-wide N tile
    const int half   = lane >> 4;   // lane half selects K/M sub-position

    // Wave-uniform tile id (scalar loop => EXEC stays all-ones around WMMA).
    int wid = (int)(blockIdx.x * (blockDim.x >> 5)) + (threadIdx.x >> 5);
    wid = __builtin_amdgcn_readfirstlane(wid);
    const int total_waves = (int)(gridDim.x * (blockDim.x >> 5));

    // Per-lane base into the LDS gate matrix: A operand for (m,k) is the
    // 8-byte pair lgate[(16m + lane_n)*36 + 4k + 2*half .. +1].
    const float* __restrict__ lg = lgate + lane_n * 36 + 2 * half;

    // Per-lane element offsets for the B-operand (X) gathers:
    // row ab = 4k + j + 2*half  ->  (ab/6)*6^7 + (ab%6)*6^4, plus lane's column.
    int xoff[9][2];
#pragma unroll
    for (int k = 0; k < 9; ++k) {
#pragma unroll
        for (int j = 0; j < 2; ++j) {
            const int ab = 4 * k + j + 2 * half;
            xoff[k][j] = (ab / 6) * C6_7 + (ab % 6) * C6_4 + lane_n;
        }
    }

    // Per-lane element offsets for C/D stores: row AB = 16m + v + 8*half.
    int soff[3][8];
#pragma unroll
    for (int m = 0; m < 3; ++m) {
#pragma unroll
        for (int v = 0; v < 8; ++v) {
            const int AB = 16 * m + v + 8 * half;
            soff[m][v] = (AB < 36) ? ((AB / 6) * C6_7 + (AB % 6) * C6_4 + lane_n) : 0;
        }
    }

    // ---- Grid-stride loop over 16-column tiles ----
    for (int t = wid; t < NUM_TILES; t += total_waves) {
        const int ro   = t / 81;          // rest_outer flat index (i0,i1,i3,i4)
        const int col0 = (t % 81) << 4;   // rest_inner start (contiguous)
        int r = ro;
        const int i4 = r % 6; r /= 6;
        const int i3 = r % 6; r /= 6;
        const int i1 = r % 6;
        const int i0 = r / 6;
        const long base = (long)i0 * C6_9 + (long)i1 * C6_8 +
                          (long)i3 * C6_6 + (long)i4 * C6_5 + col0;
        const float* __restrict__ xp = X + base;
        float* __restrict__ op = Out + base;

        // Load all 9 K-step B-operands up front (memory-level parallelism;
        // each half-wave load is 64 B contiguous).
        v2f bx[9];
#pragma unroll
        for (int k = 0; k < 9; ++k) {
            bx[k][0] = xp[xoff[k][0]];
            bx[k][1] = xp[xoff[k][1]];
        }

        v8f acc0 = {}, acc1 = {}, acc2 = {};
#pragma unroll
        for (int k = 0; k < 9; ++k) {
            // A operands from LDS (row stride 36 dwords across 16 lanes:
            // gcd(36,64)=4 -> 16 distinct banks, conflict-free).
            const v2f a0 = *(const v2f*)(lg + 0   * 36 + 4 * k);
            const v2f a1 = *(const v2f*)(lg + 16  * 36 + 4 * k);
            const v2f a2 = *(const v2f*)(lg + 32  * 36 + 4 * k);
            acc0 = __builtin_amdgcn_wmma_f32_16x16x4_f32(
                false, a0, false, bx[k], (short)0, acc0, false, false);
            acc1 = __builtin_amdgcn_wmma_f32_16x16x4_f32(
                false, a1, false, bx[k], (short)0, acc1, false, false);
            acc2 = __builtin_amdgcn_wmma_f32_16x16x4_f32(
                false, a2, false, bx[k], (short)0, acc2, false, false);
        }

        // Stores: rows 0..31 fully valid; rows 32..35 only when half==0 && v<4.
#pragma unroll
        for (int v = 0; v < 8; ++v) op[soff[0][v]] = acc0[v];
#pragma unroll
        for (int v = 0; v < 8; ++v) op[soff[1][v]] = acc1[v];
#pragma unroll
        for (int v = 0; v < 8; ++v) {
            const int AB = 32 + v + 8 * half;
            if (AB < 36) op[soff[2][v]] = acc2[v];
        }
    }
}

extern "C" void kernel_launch(void* const* d_in, const int* in_sizes, int n_in,
                              void* d_out, int out_size, void* d_ws, size_t ws_size,
                              hipStream_t stream) {
    const float* x  = (const float*)d_in[0];  // 6^10 f32
    const float* Bm = (const float*)d_in[1];  // 36*36 f32 (row-major view of (6,6,6,6))
    float* out = (float*)d_out;               // 6^10 f32

    // 2048 blocks * 4 waves = 8192 waves -> ~12.8 tiles/wave; grid-stride handles rest.
    dim3 grid(2048), block(128);
    gate_wmma_kernel<<<grid, block, 0, stream>>>(x, Bm, out);
}